// CSGat_79078937853962
// MI455X (gfx1250) — compile-verified
//
#include <hip/hip_runtime.h>
#include <hip/hip_bf16.h>
#include <math.h>

typedef __attribute__((ext_vector_type(2))) float v2f;
typedef __attribute__((ext_vector_type(8))) float v8f;

#define HEADS 4
#define OUT_DIM 64
#define FDIM 256          // HEADS * OUT_DIM
#define NEG_SLOPE 0.2f

// ---------------------------------------------------------------------------
// fp32 WMMA GEMM: C[M x Nc] = A[M x K] * B[K x Nc], all row-major, Nc == 256.
// 512 threads = 16 waves per block; wave w computes the 16x16 tile at
// (blockIdx.x*16, w*16). K stepped by 4 via V_WMMA_F32_16X16X4_F32.
//
// A fragment (16x4 f32, 2 VGPR): lanes 0-15 hold K=k+0/k+1, lanes 16-31 hold
// K=k+2/k+3, row = lane&15. B fragment (4x16): same K split, col = lane&15.
// C/D (16x16 f32, 8 VGPR): lane group selects M base (0 or 8), col = lane&15.
// ---------------------------------------------------------------------------
__global__ __launch_bounds__(512)
void gat_gemm_wmma_f32(const float* __restrict__ A, const float* __restrict__ B,
                       float* __restrict__ C, int M, int K) {
    const int wave = threadIdx.x >> 5;
    const int lane = threadIdx.x & 31;
    const int row0 = blockIdx.x * 16;
    if (row0 >= M) return;
    const int col0  = wave * 16;          // 16 waves cover all 256 columns
    const int r     = lane & 15;          // row (for A) / col (for B,C)
    const int khalf = (lane >> 4) * 2;    // 0 or 2

    const float* __restrict__ Arow = A + (size_t)(row0 + r) * K;

    v8f acc = {};
    for (int k = 0; k < K; k += 4) {
        v2f a, b;
        a.x = Arow[k + khalf];
        a.y = Arow[k + khalf + 1];
        b.x = B[(size_t)(k + khalf)     * FDIM + col0 + r];
        b.y = B[(size_t)(k + khalf + 1) * FDIM + col0 + r];
        acc = __builtin_amdgcn_wmma_f32_16x16x4_f32(
            /*neg_a=*/false, a, /*neg_b=*/false, b,
            /*c_mod=*/(short)0, acc, /*reuse_a=*/false, /*reuse_b=*/false);
    }

    const int rowc = row0 + (lane >> 4) * 8;
    const int cc   = col0 + r;
#pragma unroll
    for (int i = 0; i < 8; ++i)
        C[(size_t)(rowc + i) * FDIM + cc] = acc[i];
}

// ---------------------------------------------------------------------------
// el[n,h] = dot(feat[n,h,:], al[h,:]); er likewise. One thread per (n,h).
// ---------------------------------------------------------------------------
__global__ __launch_bounds__(256)
void gat_eler(const float* __restrict__ feat, const float* __restrict__ al,
              const float* __restrict__ ar, float* __restrict__ el,
              float* __restrict__ er, int N) {
    int idx = blockIdx.x * blockDim.x + threadIdx.x;
    if (idx >= N * HEADS) return;
    const int n = idx >> 2, h = idx & 3;
    const float* __restrict__ f = feat + (size_t)n * FDIM + h * OUT_DIM;
    const float* __restrict__ a = al + h * OUT_DIM;
    const float* __restrict__ b = ar + h * OUT_DIM;
    float sl = 0.f, sr = 0.f;
#pragma unroll 8
    for (int d = 0; d < OUT_DIM; ++d) { float v = f[d]; sl += v * a[d]; sr += v * b[d]; }
    el[idx] = sl;
    er[idx] = sr;
}

// out[n,c] = bias[c] (+ resid[n,c] if resid != nullptr)
__global__ __launch_bounds__(256)
void gat_init_out(float* __restrict__ out, const float* __restrict__ bias,
                  const float* __restrict__ resid, int N) {
    int idx = blockIdx.x * blockDim.x + threadIdx.x;
    if (idx >= N * FDIM) return;
    float v = bias[idx & (FDIM - 1)];
    if (resid) v += resid[idx];
    out[idx] = v;
}

__global__ __launch_bounds__(256)
void gat_init_maxden(float* __restrict__ emax, float* __restrict__ den, int n4) {
    int idx = blockIdx.x * blockDim.x + threadIdx.x;
    if (idx >= n4) return;
    emax[idx] = -INFINITY;
    den[idx]  = 0.f;
}

__device__ __forceinline__ void atomicMaxF(float* addr, float val) {
    // Monotonic-bits trick: valid because init is -inf.
    if (val >= 0.f)
        atomicMax((int*)addr, __float_as_int(val));
    else
        atomicMin((unsigned int*)addr, __float_as_uint(val));
}

// Pass 1: e = leaky_relu(el[src]+er[dst]); segment-max into emax[dst].
__global__ __launch_bounds__(256)
void gat_edge_logits(const int* __restrict__ src, const int* __restrict__ dst,
                     const float* __restrict__ el, const float* __restrict__ er,
                     float* __restrict__ ebuf, float* __restrict__ emax, int E) {
    int idx = blockIdx.x * blockDim.x + threadIdx.x;
    if (idx >= E * HEADS) return;
    const int e = idx >> 2, h = idx & 3;
    const int s = src[e], d = dst[e];
    float v = el[s * HEADS + h] + er[d * HEADS + h];
    v = (v >= 0.f) ? v : NEG_SLOPE * v;
    ebuf[idx] = v;
    atomicMaxF(&emax[d * HEADS + h], v);
}

// Pass 2: ee = exp(e - emax[dst]); segment-sum into denom[dst].
__global__ __launch_bounds__(256)
void gat_edge_exp(const int* __restrict__ dst, float* __restrict__ ebuf,
                  const float* __restrict__ emax, float* __restrict__ den, int E) {
    int idx = blockIdx.x * blockDim.x + threadIdx.x;
    if (idx >= E * HEADS) return;
    const int e = idx >> 2, h = idx & 3;
    const int d = dst[e];
    float ee = __expf(ebuf[idx] - emax[d * HEADS + h]);
    ebuf[idx] = ee;
    atomicAdd(&den[d * HEADS + h], ee);
}

// Pass 3: out[dst, c] += feat[src, c] * (ee / denom[dst, h]).
// One 256-thread block per edge; thread t owns channel t (h = t>>6).
// feat + out live in the 192MB L2, so this is L2-bound gather/scatter.
__global__ __launch_bounds__(256)
void gat_edge_aggregate(const int* __restrict__ src, const int* __restrict__ dst,
                        const float* __restrict__ feat, const float* __restrict__ ebuf,
                        const float* __restrict__ den, float* __restrict__ out, int E) {
    const int t = threadIdx.x;          // 0..255
    const int h = t >> 6;
    for (int e = blockIdx.x; e < E; e += gridDim.x) {
        const int s = src[e], d = dst[e];
        const float alpha = ebuf[e * HEADS + h] / den[d * HEADS + h];
        atomicAdd(&out[(size_t)d * FDIM + t], feat[(size_t)s * FDIM + t] * alpha);
    }
}

// ---------------------------------------------------------------------------
// Orchestration: two GAT layers. Input order:
// x, w1, b1, al1, ar1, w2, b2, al2, ar2, src, dst
// ---------------------------------------------------------------------------
extern "C" void kernel_launch(void* const* d_in, const int* in_sizes, int n_in,
                              void* d_out, int out_size, void* d_ws, size_t ws_size,
                              hipStream_t stream) {
    const float* x   = (const float*)d_in[0];
    const float* w1  = (const float*)d_in[1];
    const float* b1  = (const float*)d_in[2];
    const float* al1 = (const float*)d_in[3];
    const float* ar1 = (const float*)d_in[4];
    const float* w2  = (const float*)d_in[5];
    const float* b2  = (const float*)d_in[6];
    const float* al2 = (const float*)d_in[7];
    const float* ar2 = (const float*)d_in[8];
    const int*   src = (const int*)d_in[9];
    const int*   dst = (const int*)d_in[10];

    const int N = in_sizes[0] / OUT_DIM;   // 50000
    const int E = in_sizes[9];             // 850000

    // Workspace layout (floats)
    float* ws   = (float*)d_ws;
    float* feat = ws;                               // N*256 (reused both layers)
    float* h1   = feat + (size_t)N * FDIM;          // N*256 (layer-1 output)
    float* ebuf = h1   + (size_t)N * FDIM;          // E*4
    float* el   = ebuf + (size_t)E * HEADS;         // N*4
    float* er   = el   + (size_t)N * HEADS;         // N*4
    float* emax = er   + (size_t)N * HEADS;         // N*4
    float* den  = emax + (size_t)N * HEADS;         // N*4
    float* outp = (float*)d_out;                    // N*256

    const int rowTiles = (N + 15) / 16;
    const int nhBlocks = (N * HEADS + 255) / 256;
    const int ncBlocks = (N * FDIM + 255) / 256;
    const int ehBlocks = (E * HEADS + 255) / 256;

    // ---------------- Layer 1 (no residual) ----------------
    gat_gemm_wmma_f32<<<rowTiles, 512, 0, stream>>>(x, w1, feat, N, OUT_DIM);
    gat_eler<<<nhBlocks, 256, 0, stream>>>(feat, al1, ar1, el, er, N);
    gat_init_out<<<ncBlocks, 256, 0, stream>>>(h1, b1, nullptr, N);
    gat_init_maxden<<<nhBlocks, 256, 0, stream>>>(emax, den, N * HEADS);
    gat_edge_logits<<<ehBlocks, 256, 0, stream>>>(src, dst, el, er, ebuf, emax, E);
    gat_edge_exp<<<ehBlocks, 256, 0, stream>>>(dst, ebuf, emax, den, E);
    gat_edge_aggregate<<<E, 256, 0, stream>>>(src, dst, feat, ebuf, den, h1, E);

    // ---------------- Layer 2 (identity residual into d_out) ----------------
    gat_gemm_wmma_f32<<<rowTiles, 512, 0, stream>>>(h1, w2, feat, N, FDIM);
    gat_eler<<<nhBlocks, 256, 0, stream>>>(feat, al2, ar2, el, er, N);
    gat_init_out<<<ncBlocks, 256, 0, stream>>>(outp, b2, h1, N);
    gat_init_maxden<<<nhBlocks, 256, 0, stream>>>(emax, den, N * HEADS);
    gat_edge_logits<<<ehBlocks, 256, 0, stream>>>(src, dst, el, er, ebuf, emax, E);
    gat_edge_exp<<<ehBlocks, 256, 0, stream>>>(dst, ebuf, emax, den, E);
    gat_edge_aggregate<<<E, 256, 0, stream>>>(src, dst, feat, ebuf, den, outp, E);
}